// GCNEncoder_74191265071481
// MI455X (gfx1250) — compile-verified
//
#include <hip/hip_runtime.h>

typedef float v2f __attribute__((ext_vector_type(2)));
typedef float v8f __attribute__((ext_vector_type(8)));

#define N_NODES 100000
#define N_EDGES 800000
#define FEAT 256          // output width of both GCN layers
#define PSTRIDE 288       // LDS floats per K-row-pair: 2*128 + 32 pad
                          // (288 % 64 == 32 -> the two lane-halves use disjoint bank halves)

// ---------------- degree / normalization ----------------
__global__ void k_set1(float* deg, int n) {
  int i = blockIdx.x * blockDim.x + threadIdx.x;
  if (i < n) deg[i] = 1.0f;                    // self-loop contribution
}

__global__ void k_deg(const int* __restrict__ dst, float* deg, int ne) {
  int e = blockIdx.x * blockDim.x + threadIdx.x;
  if (e < ne) unsafeAtomicAdd(&deg[dst[e]], 1.0f);
}

__global__ void k_rsqrt(float* dinv, int n) {
  int i = blockIdx.x * blockDim.x + threadIdx.x;
  if (i < n) dinv[i] = rsqrtf(dinv[i]);        // deg >= 1 always (self loops)
}

// ---- f32 WMMA GEMM with fused self-loop epilogue ----
// C [MxFEAT] = A[MxK] @ B[KxFEAT];  Cs[m] = dinv[m]^2 * C[m]  (agg init)
// block = 256 threads (8 waves); block tile 128 rows x 128 cols.
// Weight panel staged in LDS, K-pair interleaved: Bs[p*PSTRIDE + 2*col + (k&1)]
// with p = k/2, so a lane's B fragment {B[k'][col], B[k'+1][col]} is one
// contiguous 8B element -> ds_load_b64 / ds_load_2addr_b64 straight into WMMA.
// K-loop is software-pipelined: fragments for step k+4 load while step k's
// 8 WMMAs execute (double-buffered registers), hiding LDS/global latency.
//
// Fragment layouts (ISA 7.12.2, V_WMMA_F32_16X16X4_F32):
//   A 16x4 : lane-half kh selects K-half, VGPR j -> a[j] = A[row][k + 2*kh + j]
//   B 4x16 : b[j] = B[k + 2*kh + j][col]  -> pair index p = k/2 + kh
//   C/D    : VGPR v -> row (v + 8*kh), lane&15 -> col
__global__ __launch_bounds__(256)
void k_gemm_f32(const float* __restrict__ A, const float* __restrict__ B,
                float* __restrict__ C, float* __restrict__ Cs,
                const float* __restrict__ dinv, int M, int K) {
  extern __shared__ float Bs[];                 // (K/2) * PSTRIDE floats

  const int lane = threadIdx.x & 31;
  const int wave = threadIdx.x >> 5;
  const int l15  = lane & 15;
  const int kh   = lane >> 4;                   // K-half select (0/1)
  const int row0 = blockIdx.y * 128 + wave * 16;
  const int col0 = blockIdx.x * 128;

  // --- stage weight panel to LDS, pair-interleaved ---
  // 128 threads per row-pair: thread loads B[2p][c], B[2p+1][c] (coalesced b32)
  // and stores one ds_store_b64 at Bs[p*PSTRIDE + 2c].
  {
    const int c  = threadIdx.x & 127;
    const int ph = threadIdx.x >> 7;            // 0 or 1
    const float* Bg = B + col0 + c;
    for (int p = ph; p < (K >> 1); p += 2) {
      float lo = Bg[(size_t)(2 * p) * FEAT];
      float hi = Bg[(size_t)(2 * p + 1) * FEAT];
      v2f pr; pr.x = lo; pr.y = hi;
      *(v2f*)&Bs[p * PSTRIDE + 2 * c] = pr;
    }
  }
  __syncthreads();

  int r  = row0 + l15;
  int rA = r < M ? r : (M - 1);                 // clamp tail rows (stores guarded)
  const float* Ap  = A + (size_t)rA * K + 2 * kh;
  const float* Bsp = Bs + kh * PSTRIDE + 2 * l15;

  v8f acc[8] = {};
  v2f aCur, bCur[8];
  v2f aNxt, bNxt[8];

  // prologue: load fragments for k = 0
  aCur.x = Ap[0];
  aCur.y = Ap[1];
#pragma unroll
  for (int t = 0; t < 8; ++t) bCur[t] = *(const v2f*)(Bsp + t * 32);

  for (int k = 0; k < K - 4; k += 4) {
    // prefetch next K-step while current WMMAs run
    const float* Bn = Bsp + ((k + 4) >> 1) * PSTRIDE;
    aNxt.x = Ap[k + 4];
    aNxt.y = Ap[k + 5];
#pragma unroll
    for (int t = 0; t < 8; ++t) bNxt[t] = *(const v2f*)(Bn + t * 32);

#pragma unroll
    for (int t = 0; t < 8; ++t) {
      acc[t] = __builtin_amdgcn_wmma_f32_16x16x4_f32(
          false, aCur, false, bCur[t], (short)0, acc[t], false, false);
    }
    aCur = aNxt;
#pragma unroll
    for (int t = 0; t < 8; ++t) bCur[t] = bNxt[t];
  }
  // epilogue K-step
#pragma unroll
  for (int t = 0; t < 8; ++t) {
    acc[t] = __builtin_amdgcn_wmma_f32_16x16x4_f32(
        false, aCur, false, bCur[t], (short)0, acc[t], false, false);
  }

  // --- epilogue: write xw and dinv^2 * xw (self-loop agg init) ---
  const int mBase = row0 + 8 * kh;
#pragma unroll
  for (int v = 0; v < 8; ++v) {
    int m = mBase + v;
    if (m < M) {
      float dv = dinv[m];
      float s  = dv * dv;
#pragma unroll
      for (int t = 0; t < 8; ++t) {
        int c = col0 + t * 16 + l15;
        float val = acc[t][v];
        C[(size_t)m * FEAT + c]  = val;
        Cs[(size_t)m * FEAT + c] = s * val;
      }
    }
  }
}

// ---------------- edge scatter: out[dst] += dinv[s]*dinv[d] * xw[src] ----------------
// one wave per edge; lane l handles features l, l+32, ..., l+224 (coalesced)
__global__ __launch_bounds__(256)
void k_scatter(const float* __restrict__ xw, const float* __restrict__ dinv,
               const int* __restrict__ src, const int* __restrict__ dst,
               float* __restrict__ out, int ne) {
  int e = blockIdx.x * (blockDim.x >> 5) + (threadIdx.x >> 5);
  if (e >= ne) return;
  int lane = threadIdx.x & 31;
  int s = src[e], d = dst[e];
  float nrm = dinv[s] * dinv[d];
  const float* xs = xw + (size_t)s * FEAT;
  float* od = out + (size_t)d * FEAT;
#pragma unroll
  for (int i = 0; i < 8; ++i) {
    int f = lane + i * 32;
    unsafeAtomicAdd(&od[f], nrm * xs[f]);
  }
}

// ---------------- bias (+optional ReLU), vectorized float4 ----------------
__global__ void k_bias_act(const float4* __restrict__ in,
                           const float4* __restrict__ bias,
                           float4* __restrict__ out, int n4, int relu) {
  int idx = blockIdx.x * blockDim.x + threadIdx.x;
  if (idx >= n4) return;
  float4 v = in[idx];
  float4 b = bias[idx & 63];                    // 64 float4 per 256-wide row
  float4 r = make_float4(v.x + b.x, v.y + b.y, v.z + b.z, v.w + b.w);
  if (relu) {
    r.x = fmaxf(r.x, 0.0f); r.y = fmaxf(r.y, 0.0f);
    r.z = fmaxf(r.z, 0.0f); r.w = fmaxf(r.w, 0.0f);
  }
  out[idx] = r;
}

extern "C" void kernel_launch(void* const* d_in, const int* in_sizes, int n_in,
                              void* d_out, int out_size, void* d_ws, size_t ws_size,
                              hipStream_t stream) {
  const float* x  = (const float*)d_in[0];
  const int*   ei = (const int*)d_in[1];        // [2, E]: row0 = src, row1 = dst
  const float* W1 = (const float*)d_in[2];
  const float* b1 = (const float*)d_in[3];
  const float* W2 = (const float*)d_in[4];
  const float* b2 = (const float*)d_in[5];
  const int* src = ei;
  const int* dst = ei + N_EDGES;

  char* ws = (char*)d_ws;
  float* dinv = (float*)ws;                                  // N floats
  float* bufA = (float*)(ws + (1u << 20));                   // N*256 f32 (102.4 MB)
  float* bufB = bufA + (size_t)N_NODES * FEAT;               // N*256 f32 (102.4 MB)
  float* outp = (float*)d_out;

  const int n4 = N_NODES * (FEAT / 4);
  dim3 gGemm(FEAT / 128, (N_NODES + 127) / 128);
  int gElem  = (n4 + 255) / 256;
  int gNode  = (N_NODES + 255) / 256;
  int gEdgeT = (N_EDGES + 255) / 256;
  int gEdgeW = (N_EDGES + 7) / 8;               // 8 waves (edges) per block
  size_t smem1 = (size_t)(128 / 2) * PSTRIDE * 4;  // K=128:  ~72 KB
  size_t smem2 = (size_t)(256 / 2) * PSTRIDE * 4;  // K=256: ~144 KB (<320 KB WGP LDS)

  // normalization: deg = 1 + indegree, dinv = rsqrt(deg)
  k_set1<<<gNode, 256, 0, stream>>>(dinv, N_NODES);
  k_deg<<<gEdgeT, 256, 0, stream>>>(dst, dinv, N_EDGES);
  k_rsqrt<<<gNode, 256, 0, stream>>>(dinv, N_NODES);

  // layer 1: xw = x@W1 (bufA) with fused agg-init (bufB = dinv^2*xw);
  //          scatter edges into bufB; h = relu(bufB + b1) -> bufA
  k_gemm_f32<<<gGemm, 256, smem1, stream>>>(x, W1, bufA, bufB, dinv, N_NODES, 128);
  k_scatter<<<gEdgeW, 256, 0, stream>>>(bufA, dinv, src, dst, bufB, N_EDGES);
  k_bias_act<<<gElem, 256, 0, stream>>>((const float4*)bufB, (const float4*)b1,
                                        (float4*)bufA, n4, 1);

  // layer 2: hw = h@W2 (bufB) with fused agg-init directly into d_out;
  //          scatter edges into d_out; out += b2
  k_gemm_f32<<<gGemm, 256, smem2, stream>>>(bufA, W2, bufB, outp, dinv, N_NODES, 256);
  k_scatter<<<gEdgeW, 256, 0, stream>>>(bufB, dinv, src, dst, outp, N_EDGES);
  k_bias_act<<<gElem, 256, 0, stream>>>((const float4*)outp, (const float4*)b2,
                                        (float4*)outp, n4, 0);
}